// image2patch_4801773436970
// MI455X (gfx1250) — compile-verified
//
#include <hip/hip_runtime.h>
#include <cstdint>

// image2patch: input (32,8,256,256) f32 -> output (256, 3969, 64) f32
// WINDOW = 249, patch positions per axis = 63 (stride 4 covers 0..248, 248 == WINDOW-1)
#define IMG 256
#define NPOS 63
#define NPATCH (NPOS * NPOS)     // 3969
#define PSIZE 8
#define PSTRIDE 4
#define ROWPAD 264               // padded LDS row stride (floats): 264 % 64 banks = 8 -> conflict-free patch reads

typedef float v4f __attribute__((ext_vector_type(4)));

__global__ __launch_bounds__(256)
void image2patch_kernel(const float* __restrict__ x, float* __restrict__ out)
{
    __shared__ float tile[PSIZE * ROWPAD];   // 8448 bytes

    const int block = blockIdx.x;
    const int bc = block / NPOS;             // 0..255 (B*C)
    const int pr = block % NPOS;             // strip (patch-row) index

    // Strip = 8 contiguous image rows starting at row pr*4 (uniform -> lands in SGPRs)
    const float* src = x + (size_t)bc * (IMG * IMG) + (size_t)(pr * PSTRIDE) * IMG;

    const int t = threadIdx.x;
    const unsigned lds_base = (unsigned)(uintptr_t)(void*)&tile[0];

    // Stage 8 rows x 256 floats (512 x float4) into padded LDS rows via CDNA5
    // async global->LDS copies (ASYNCcnt-tracked, per-lane 16B transfers).
    #pragma unroll
    for (int i = 0; i < 2; ++i) {
        const int e   = t + i * 256;         // 0..511: float4 chunk id
        const int row = e >> 6;              // 0..7
        const int cc  = e & 63;              // float4 within row
        const unsigned lds_addr = lds_base + (unsigned)(row * ROWPAD + cc * 4) * 4u;
        const unsigned goff     = (unsigned)(row * IMG + cc * 4) * 4u;
        asm volatile("global_load_async_to_lds_b128 %0, %1, %2"
                     :: "v"(lds_addr), "v"(goff), "s"(src)
                     : "memory");
    }
    asm volatile("s_wait_asynccnt 0" ::: "memory");
    __syncthreads();

    // Output for this strip is contiguous: 63 patches * 64 floats = 1008 float4.
    // float4 q within a patch covers k = 4q..4q+3 -> di = q>>1, dj = (q&1)*4,
    // i.e. 4 contiguous cols of one row: single aligned LDS b128 read.
    float* dst = out + (size_t)bc * (NPATCH * 64) + (size_t)(pr * NPOS) * 64;
    #pragma unroll
    for (int i = 0; i < 4; ++i) {
        const int o = t + i * 256;
        if (o < NPOS * 16) {
            const int pc = o >> 4;           // patch column 0..62
            const int q  = o & 15;
            const int di = q >> 1;
            const int dj = (q & 1) * 4;
            const v4f val = *(const v4f*)&tile[di * ROWPAD + pc * PSTRIDE + dj];
            // NT store: 260 MB output stream must not evict the 64 MB input from L2
            __builtin_nontemporal_store(val, (v4f*)(dst + (size_t)o * 4));
        }
    }
}

extern "C" void kernel_launch(void* const* d_in, const int* in_sizes, int n_in,
                              void* d_out, int out_size, void* d_ws, size_t ws_size,
                              hipStream_t stream)
{
    (void)in_sizes; (void)n_in; (void)out_size; (void)d_ws; (void)ws_size;
    const float* x = (const float*)d_in[0];
    float* out = (float*)d_out;
    // 256 (B*C) * 63 strips = 16128 workgroups, 256 threads (8 wave32s) each
    image2patch_kernel<<<dim3(256 * NPOS), dim3(256), 0, stream>>>(x, out);
}